// Crossings_2791728743047
// MI455X (gfx1250) — compile-verified
//
#include <hip/hip_runtime.h>

#define NBINS 128
#define EPSI 1e-5f

// ---------------------------------------------------------------------------
// Zero the 128-bin output (harness poisons d_out with 0xAA before timing).
// ---------------------------------------------------------------------------
__global__ void Crossings_zero_kernel(float* __restrict__ out) {
    int i = blockIdx.x * blockDim.x + threadIdx.x;
    if (i < NBINS) out[i] = 0.0f;
}

// Wave-relative LDS byte offset of a __shared__ object: generic (flat) LDS
// pointers carry the LDS offset in their low 32 bits.
__device__ __forceinline__ unsigned lds_off(const void* p) {
    return (unsigned)(unsigned long long)p;
}

// Issue 4 async 16B/lane streaming loads (one per index stream) into this
// wave's LDS slice. Tracked by ASYNCcnt; completes in order. Non-temporal:
// the index streams are read exactly once, so keep them out of L2 and
// preserve residency for the node_pos/batch_index gather working set.
__device__ __forceinline__ void issue_chunk4(unsigned l0,
                                             const int* a0, const int* a1,
                                             const int* a2, const int* a3) {
    asm volatile(
        "global_load_async_to_lds_b128 %0, %4, off th:TH_LOAD_NT\n\t"
        "global_load_async_to_lds_b128 %1, %5, off th:TH_LOAD_NT\n\t"
        "global_load_async_to_lds_b128 %2, %6, off th:TH_LOAD_NT\n\t"
        "global_load_async_to_lds_b128 %3, %7, off th:TH_LOAD_NT"
        :
        : "v"(l0), "v"(l0 + 512u), "v"(l0 + 1024u), "v"(l0 + 1536u),
          "v"(a0), "v"(a1), "v"(a2), "v"(a3)
        : "memory");
}

// Proper-intersection orientation test, matches reference _edges_intersect.
// p1=np[a1], p2=np[b1], p3=np[a2], p4=np[b2]
__device__ __forceinline__ bool seg_cross(const float2* __restrict__ np,
                                          int a1, int b1, int a2, int b2) {
    float2 p1 = np[a1];
    float2 p2 = np[b1];
    float2 p3 = np[a2];
    float2 p4 = np[b2];
    float ax = p4.x - p3.x, ay = p4.y - p3.y;           // p4 - p3
    float d1 = ax * (p1.y - p3.y) - ay * (p1.x - p3.x); // cross(p4-p3, p1-p3)
    float d2 = ax * (p2.y - p3.y) - ay * (p2.x - p3.x); // cross(p4-p3, p2-p3)
    float bx = p2.x - p1.x, by = p2.y - p1.y;           // p2 - p1
    float d3 = bx * (p3.y - p1.y) - by * (p3.x - p1.x); // cross(p2-p1, p3-p1)
    float d4 = bx * (p4.y - p1.y) - by * (p4.x - p1.x); // cross(p2-p1, p4-p1)
    return (d1 * d2 < -EPSI) && (d3 * d4 < -EPSI);
}

// ---------------------------------------------------------------------------
// Main kernel: 256 threads = 8 waves. Each wave double-buffers 128-pair
// chunks of the 4 index streams through its own 4KB LDS slice via async
// global->LDS loads, tests crossings, and LDS-atomically histograms into
// 128 bins; per-block partials are float-atomically added to d_out.
// ---------------------------------------------------------------------------
__global__ __launch_bounds__(256) void Crossings_main_kernel(
    const float2* __restrict__ npos,
    const int*    __restrict__ batch,
    const int*    __restrict__ epi,
    long long P, long long Pfull,
    float* __restrict__ out)
{
    __shared__ unsigned bins[NBINS];
    __shared__ __align__(16) int sbuf[8192];  // 8 waves * 2 buffers * 4 streams * 32 lanes * 16B

    for (int i = threadIdx.x; i < NBINS; i += blockDim.x) bins[i] = 0u;
    __syncthreads();

    const int lane = threadIdx.x & 31;
    const int wv   = threadIdx.x >> 5;
    const long long nw = (long long)gridDim.x * (blockDim.x >> 5);
    const long long gw = (long long)blockIdx.x * (blockDim.x >> 5) + wv;

    const int* s1p = epi;
    const int* s2p = epi + P;
    const int* e1p = epi + 2 * P;
    const int* e2p = epi + 3 * P;

    // This lane's base inside its wave's LDS slice (4KB per wave, 16B per lane).
    const char* mybuf = (const char*)sbuf + (wv << 12) + (lane << 4);
    const unsigned myoff = lds_off(mybuf);

    long long       base   = gw * 128;
    const long long stride = nw * 128;
    bool have = (base + 128) <= Pfull;
    if (have) {
        long long q = base + (long long)(lane << 2);
        issue_chunk4(myoff, s1p + q, s2p + q, e1p + q, e2p + q);
    }
    int bsel = 0;
    while (have) {
        long long nxt = base + stride;
        bool havenext = (nxt + 128) <= Pfull;
        if (havenext) {
            long long q = nxt + (long long)(lane << 2);
            issue_chunk4(myoff + (unsigned)((bsel ^ 1) << 11),
                         s1p + q, s2p + q, e1p + q, e2p + q);
            asm volatile("s_wait_asynccnt 4" ::: "memory");  // current chunk landed
        } else {
            asm volatile("s_wait_asynccnt 0" ::: "memory");
        }

        const char* cb = mybuf + (bsel << 11);
        int4 vs1 = *(const int4*)(cb);
        int4 vs2 = *(const int4*)(cb + 512);
        int4 ve1 = *(const int4*)(cb + 1024);
        int4 ve2 = *(const int4*)(cb + 1536);

        int s1a[4] = {vs1.x, vs1.y, vs1.z, vs1.w};
        int s2a[4] = {vs2.x, vs2.y, vs2.z, vs2.w};
        int e1a[4] = {ve1.x, ve1.y, ve1.z, ve1.w};
        int e2a[4] = {ve2.x, ve2.y, ve2.z, ve2.w};

#pragma unroll
        for (int j = 0; j < 4; ++j) {
            if (seg_cross(npos, s1a[j], e1a[j], s2a[j], e2a[j])) {
                atomicAdd(&bins[batch[s1a[j]]], 1u);
            }
        }

        base = nxt;
        bsel ^= 1;
        have = havenext;
    }

    // Tail (and misaligned-P fallback): plain guarded loads for p in [Pfull, P).
    {
        const long long tid     = (long long)blockIdx.x * blockDim.x + threadIdx.x;
        const long long tstride = (long long)gridDim.x * blockDim.x;
        for (long long p = Pfull + tid; p < P; p += tstride) {
            int i1 = s1p[p], i2 = s2p[p], i3 = e1p[p], i4 = e2p[p];
            if (seg_cross(npos, i1, i3, i2, i4)) {
                atomicAdd(&bins[batch[i1]], 1u);
            }
        }
    }

    __syncthreads();
    for (int i = threadIdx.x; i < NBINS; i += blockDim.x) {
        unsigned c = bins[i];
        if (c) atomicAdd(&out[i], (float)c);  // integer-valued: exact, order-free
    }
}

extern "C" void kernel_launch(void* const* d_in, const int* in_sizes, int n_in,
                              void* d_out, int out_size, void* d_ws, size_t ws_size,
                              hipStream_t stream) {
    (void)n_in; (void)d_ws; (void)ws_size; (void)out_size;

    const float2* npos  = (const float2*)d_in[0];
    const int*    batch = (const int*)d_in[3];
    const int*    epi   = (const int*)d_in[4];  // [4][P]: s1, s2, e1, e2

    const long long P = (long long)in_sizes[4] / 4;
    // Async b128 path needs the four stream bases 16B-aligned (P % 4 == 0).
    const long long Pfull = ((P & 3) == 0) ? (P / 128) * 128 : 0;

    float* out = (float*)d_out;
    Crossings_zero_kernel<<<1, NBINS, 0, stream>>>(out);

    const int blocks = 2048;  // 16384 waves -> ~7.6 chunks/wave at P=16M
    Crossings_main_kernel<<<blocks, 256, 0, stream>>>(npos, batch, epi, P, Pfull, out);
}